// Attention_16355235463288
// MI455X (gfx1250) — compile-verified
//
#include <hip/hip_runtime.h>

// ---------------------------------------------------------------------------
// Problem constants (from the reference): B=32, H=W=64, D=N=C=512
// ---------------------------------------------------------------------------
constexpr int B_ = 32, H_ = 64, W_ = 64, D_ = 512, N_ = 512, C_ = 512;
constexpr int HW_ = H_ * W_;          // 4096
constexpr int LOCS = B_ * HW_;        // 131072

typedef __attribute__((ext_vector_type(2))) float v2f;
typedef __attribute__((ext_vector_type(8))) float v8f;

// gfx1250 has a hardware V_TANH_F32 transcendental; use it if the builtin
// is target-available, else fall back to libm tanhf.
#if defined(__has_builtin)
#if __has_builtin(__builtin_amdgcn_tanhf)
#define FAST_TANH(x) __builtin_amdgcn_tanhf(x)
#endif
#endif
#ifndef FAST_TANH
#define FAST_TANH(x) tanhf(x)
#endif

// ---------------------------------------------------------------------------
// Phase 1: q[b,d] = sum_n ht[b,n] * Wq[d,n]   via V_WMMA_F32_16X16X4_F32
// One wave (32 threads) per 16x16 output tile; K stepped by 4.
// A (16x4 f32): lanes 0-15 hold {K=0,K=1} in vgpr{0,1}, lanes 16-31 {K=2,K=3}
// B (4x16 f32): mirrored — lane n%16 holds column n, K rows split by lane half
// C/D (16x16 f32): vgpr r holds rows r (lanes 0-15) and r+8 (lanes 16-31)
// ---------------------------------------------------------------------------
__global__ void qproj_wmma(const float* __restrict__ ht,
                           const float* __restrict__ Wq,
                           float* __restrict__ q) {
  const int lane = threadIdx.x;          // 0..31, wave32
  const int half = lane >> 4;            // 0 or 1
  const int lm   = lane & 15;
  const int n0 = blockIdx.x * 16;        // tile over D (output cols)
  const int m0 = blockIdx.y * 16;        // tile over B (output rows)

  const float* arow = ht + (size_t)(m0 + lm) * N_;   // A row for this lane
  const float* bcol = Wq + (size_t)(n0 + lm) * N_;   // B col n = Wq row d

  v8f c = {};
  for (int k0 = 0; k0 < N_; k0 += 4) {
    const int ka = k0 + half * 2;
    v2f a = *(const v2f*)(arow + ka);    // {K=ka, K=ka+1} for row m0+lm
    v2f b = *(const v2f*)(bcol + ka);    // {K=ka, K=ka+1} for col n0+lm
    c = __builtin_amdgcn_wmma_f32_16x16x4_f32(
        /*neg_a=*/false, a, /*neg_b=*/false, b,
        /*c_mod=*/(short)0, c, /*reuse_a=*/false, /*reuse_b=*/false);
  }

  float* out = q + (size_t)(m0 + half * 8) * D_ + n0 + lm;
#pragma unroll
  for (int r = 0; r < 8; ++r) out[(size_t)r * D_] = c[r];
}

// ---------------------------------------------------------------------------
// Phase 2: s_raw[b,h,w] = sum_d tanh(key[b,h,w,d] + q[b,d]) * Wa[d] + ba
// One wave per location; 512 floats per location, float4-coalesced.
// This is the first 268 MB streaming pass (bandwidth-bound).
// ---------------------------------------------------------------------------
__global__ void score_kernel(const float* __restrict__ key,
                             const float* __restrict__ q,
                             const float* __restrict__ Wa,
                             const float* __restrict__ ba,
                             float* __restrict__ s_raw) {
  const int wave = threadIdx.x >> 5;
  const int lane = threadIdx.x & 31;
  const long loc = (long)blockIdx.x * 8 + wave;   // 0..LOCS-1
  const int  b   = (int)(loc >> 12);              // / 4096

  const float* kp = key + loc * (long)D_;
  const float* qp = q + (size_t)b * D_;

  float acc = 0.f;
#pragma unroll
  for (int i = 0; i < 4; ++i) {
    const int idx = lane * 4 + i * 128;           // 512B contiguous per step
    float4 kv = *(const float4*)(kp + idx);
    float4 qv = *(const float4*)(qp + idx);
    float4 wv = *(const float4*)(Wa + idx);
    acc += FAST_TANH(kv.x + qv.x) * wv.x;
    acc += FAST_TANH(kv.y + qv.y) * wv.y;
    acc += FAST_TANH(kv.z + qv.z) * wv.z;
    acc += FAST_TANH(kv.w + qv.w) * wv.w;
  }
#pragma unroll
  for (int off = 16; off > 0; off >>= 1) acc += __shfl_xor(acc, off, 32);
  if (lane == 0) s_raw[loc] = acc + ba[0];
}

// ---------------------------------------------------------------------------
// Phase 3: global max over all B*H*W scores (512 KB — negligible).
// ---------------------------------------------------------------------------
__global__ void max_kernel(const float* __restrict__ s_raw,
                           float* __restrict__ gmax) {
  __shared__ float sm[1024];
  float m = -3.402823466e38f;
  for (int i = threadIdx.x; i < LOCS; i += 1024) m = fmaxf(m, s_raw[i]);
  sm[threadIdx.x] = m;
  __syncthreads();
  for (int s = 512; s > 0; s >>= 1) {
    if ((int)threadIdx.x < s) sm[threadIdx.x] = fmaxf(sm[threadIdx.x], sm[threadIdx.x + s]);
    __syncthreads();
  }
  if (threadIdx.x == 0) gmax[0] = sm[0];
}

// ---------------------------------------------------------------------------
// Phase 4: per batch: p = exp(s - gmax) * mask; s_out = p / (sum(p) + 1e-10)
// One block per batch (4096 elems, 16 per thread held in registers).
// Normalized s written straight to the output's s region; phase 5 reads it.
// ---------------------------------------------------------------------------
__global__ void norm_kernel(const float* __restrict__ s_raw,
                            const float* __restrict__ mask,
                            const float* __restrict__ gmax,
                            float* __restrict__ s_out) {
  __shared__ float sm[256];
  const int b = blockIdx.x;
  const float m = gmax[0];
  const float* sr = s_raw + (size_t)b * HW_;
  const float* mk = mask + (size_t)b * HW_;

  float p[16];
  float sum = 0.f;
#pragma unroll
  for (int i = 0; i < 16; ++i) {
    const int idx = threadIdx.x + i * 256;
    const float v = expf(sr[idx] - m) * mk[idx];
    p[i] = v;
    sum += v;
  }
  sm[threadIdx.x] = sum;
  __syncthreads();
  for (int s = 128; s > 0; s >>= 1) {
    if ((int)threadIdx.x < s) sm[threadIdx.x] += sm[threadIdx.x + s];
    __syncthreads();
  }
  const float inv = 1.f / (sm[0] + 1e-10f);
  float* so = s_out + (size_t)b * HW_;
#pragma unroll
  for (int i = 0; i < 16; ++i) so[threadIdx.x + i * 256] = p[i] * inv;
}

// ---------------------------------------------------------------------------
// Phase 5: ct[b,c] = sum_hw ctx_val[b,c,hw] * s_norm[b,hw]
// One block per (b, 8-channel group): stage s_norm[b] (16 KB) in LDS once,
// then one wave per channel streams 16 KB of ctx_val (float4).
// Second 268 MB streaming pass (bandwidth-bound).
// ---------------------------------------------------------------------------
__global__ void ctx_kernel(const float* __restrict__ val,
                           const float* __restrict__ s_norm,
                           float* __restrict__ ct) {
  __shared__ __align__(16) float p_sh[HW_];   // 16 KB of the 320 KB LDS
  const int b    = blockIdx.y;
  const int wave = threadIdx.x >> 5;
  const int lane = threadIdx.x & 31;

  const float4* src = (const float4*)(s_norm + (size_t)b * HW_);
  float4* dst = (float4*)p_sh;
  for (int i = threadIdx.x; i < HW_ / 4; i += 256) dst[i] = src[i];
  __syncthreads();

  const int c = blockIdx.x * 8 + wave;
  const float* vp = val + ((size_t)b * C_ + c) * HW_;
  float acc = 0.f;
#pragma unroll 4
  for (int i = 0; i < 32; ++i) {
    const int idx = lane * 4 + i * 128;
    float4 v = *(const float4*)(vp + idx);
    float4 p = *(const float4*)(p_sh + idx);
    acc += v.x * p.x + v.y * p.y + v.z * p.z + v.w * p.w;
  }
#pragma unroll
  for (int off = 16; off > 0; off >>= 1) acc += __shfl_xor(acc, off, 32);
  if (lane == 0) ct[(size_t)b * C_ + c] = acc;
}

// ---------------------------------------------------------------------------
// Launch. Inputs (setup_inputs order):
//   0 ctx_val [B,C,H,W]  1 ctx_key [B,H,W,D]  2 ctx_mask [B,H,W]
//   3 ht_query [B,N]     4 Wq [D,N]           5 Wa [1,D]   6 ba [1]
// Output: ct [B,C] then s [B,H,W], concatenated flat (all fp32).
// ---------------------------------------------------------------------------
extern "C" void kernel_launch(void* const* d_in, const int* in_sizes, int n_in,
                              void* d_out, int out_size, void* d_ws, size_t ws_size,
                              hipStream_t stream) {
  const float* ctx_val  = (const float*)d_in[0];
  const float* ctx_key  = (const float*)d_in[1];
  const float* ctx_mask = (const float*)d_in[2];
  const float* ht       = (const float*)d_in[3];
  const float* Wq       = (const float*)d_in[4];
  const float* Wa       = (const float*)d_in[5];
  const float* ba       = (const float*)d_in[6];

  float* out   = (float*)d_out;
  float* ct    = out;               // [B, C]
  float* s_out = out + B_ * C_;     // [B, H, W] (normalized attention)

  float* ws    = (float*)d_ws;
  float* q     = ws;                // B*D   = 16384 floats
  float* s_raw = q + B_ * D_;       // B*H*W = 131072 floats
  float* gmax  = s_raw + LOCS;      // 1 float

  qproj_wmma  <<<dim3(D_ / 16, B_ / 16), 32,  0, stream>>>(ht, Wq, q);
  score_kernel<<<LOCS / 8,               256, 0, stream>>>(ctx_key, q, Wa, ba, s_raw);
  max_kernel  <<<1,                      1024, 0, stream>>>(s_raw, gmax);
  norm_kernel <<<B_,                     256, 0, stream>>>(s_raw, ctx_mask, gmax, s_out);
  ctx_kernel  <<<dim3(C_ / 8, B_),       256, 0, stream>>>(ctx_val, s_out, ct);
}